// GauntConvolution_74723841016350
// MI455X (gfx1250) — compile-verified
//
#include <hip/hip_runtime.h>
#include <hip/hip_bf16.h>

// ---- static problem constants (match reference) ----
#define N_NODES 4096
#define E_EDGES 16384
#define MULT    32
#define CX9     9     // (LX+1)^2
#define CF9     9
#define COUT9   9
#define NB      64
#define XROW    (MULT * CX9)    // 288
#define WCOLS   (MULT * 3)      // 96  = MUL*(LOUT+1)

typedef __attribute__((ext_vector_type(2))) float v2f;
typedef __attribute__((ext_vector_type(8))) float v8f;

// ShiftedSoftPlus: softplus(v) - ln(2), numerically stable form
__device__ __forceinline__ float ssp_f(float v) {
    return fmaxf(v, 0.0f) + log1pf(expf(-fabsf(v))) - 0.69314718055994531f;
}

// ---------------------------------------------------------------------------
// Kernel A: build the 9x9x9 real Gaunt coupling tensor
//   G[co,cx,cf] = (1/81) * Re( sum_{u,v} conv2_full(Mx[cx], Mf[cf])[u,v] * Ms[u,v,co] )
// Mx: (9,5,5) c64, Mf: (9,5,5) c64, Ms: (9,9,9) c64 (interleaved re,im floats)
// One block per (cx,cf) pair; 81 threads compute the 9x9 complex conv into LDS,
// 9 threads contract against Ms.
// ---------------------------------------------------------------------------
__global__ void k_gaunt_G(const float* __restrict__ Mx, const float* __restrict__ Mf,
                          const float* __restrict__ Ms, float* __restrict__ gbuf) {
    __shared__ float c2r[81];
    __shared__ float c2i[81];
    const int b  = blockIdx.x;         // 0..80
    const int cx = b / 9, cf = b % 9;
    const int t  = threadIdx.x;
    if (t < 81) {
        const int u = t / 9, v = t % 9;
        float ar = 0.f, ai = 0.f;
        for (int p = 0; p < 5; ++p) {
            const int up = u - p;
            if (up < 0 || up > 4) continue;
            for (int q = 0; q < 5; ++q) {
                const int vq = v - q;
                if (vq < 0 || vq > 4) continue;
                const int ia = ((cx * 5 + p) * 5 + q) * 2;
                const int ib = ((cf * 5 + up) * 5 + vq) * 2;
                const float xr = Mx[ia], xi = Mx[ia + 1];
                const float fr = Mf[ib], fi = Mf[ib + 1];
                ar += xr * fr - xi * fi;
                ai += xr * fi + xi * fr;
            }
        }
        c2r[t] = ar;
        c2i[t] = ai;
    }
    __syncthreads();
    if (t < 9) {
        const int co = t;
        float g = 0.f;
        for (int uv = 0; uv < 81; ++uv) {
            const int im = (uv * 9 + co) * 2;
            g += c2r[uv] * Ms[im] - c2i[uv] * Ms[im + 1];
        }
        gbuf[co * 81 + cx * 9 + cf] = g * (1.0f / 81.0f);
    }
}

// ---------------------------------------------------------------------------
// Kernel B: zero the output accumulator (harness poisons d_out)
// ---------------------------------------------------------------------------
__global__ void k_zero(float* __restrict__ out, int n) {
    const int i = blockIdx.x * blockDim.x + threadIdx.x;
    if (i < n) out[i] = 0.0f;
}

// ---------------------------------------------------------------------------
// Kernel C: radial-MLP path weights with fp32 WMMA (v_wmma_f32_16x16x4_f32)
//   h = ssp( embed(16x64) @ w1(64x64) * 1/8 )   [fused via LDS]
//   w =      h(16x64)    @ w2(64x96) * 1/8   -> wbuf (E x 96)
// Block = 192 threads (6 wave32). Block handles 16 edges.
// Phase 1: waves 0..3 each own a 16-col N-tile of h. Phase 2: all 6 waves
// each own a 16-col N-tile of the 96-wide output.
// A layout (16x4 f32): lane<16 -> M=lane, K pair {0,1}; lane>=16 -> M=lane-16, K pair {2,3}
// B layout (4x16):     lane<16 -> K rows {0,1} at N=lane; lane>=16 -> K rows {2,3} at N=lane-16
// C layout (16x16):    vgpr r -> M = r + 8*(lane>=16), N = lane%16
// ---------------------------------------------------------------------------
__global__ void __launch_bounds__(192) k_mlp(const float* __restrict__ embed,
                                             const float* __restrict__ w1,
                                             const float* __restrict__ w2,
                                             float* __restrict__ wbuf) {
    __shared__ float hbuf[16][64];
    const int tid  = threadIdx.x;
    const int wave = tid >> 5;
    const int lane = tid & 31;
    const int col  = lane & 15;
    const int hi   = lane >> 4;       // 0 or 1 (selects K pair / M half)
    const int e0   = blockIdx.x * 16;

    // ---- phase 1: GEMM1 + ShiftedSoftPlus -> LDS ----
    if (wave < 4) {
        const int n0 = wave * 16;
        v8f c = {0.f, 0.f, 0.f, 0.f, 0.f, 0.f, 0.f, 0.f};
        for (int kk = 0; kk < 16; ++kk) {
            const int kb = kk * 4 + hi * 2;
            v2f a, bf;
            a.x  = embed[(e0 + col) * NB + kb];
            a.y  = embed[(e0 + col) * NB + kb + 1];
            bf.x = w1[kb * NB + n0 + col];
            bf.y = w1[(kb + 1) * NB + n0 + col];
            c = __builtin_amdgcn_wmma_f32_16x16x4_f32(false, a, false, bf,
                                                      (short)0, c, false, false);
        }
#pragma unroll
        for (int r = 0; r < 8; ++r) {
            const int m = r + 8 * hi;
            hbuf[m][n0 + col] = ssp_f(c[r] * 0.125f);   // * 1/sqrt(NB)
        }
    }
    __syncthreads();

    // ---- phase 2: GEMM2 -> wbuf ----
    {
        const int n0 = wave * 16;
        v8f c = {0.f, 0.f, 0.f, 0.f, 0.f, 0.f, 0.f, 0.f};
        for (int kk = 0; kk < 16; ++kk) {
            const int kb = kk * 4 + hi * 2;
            v2f a, bf;
            a.x  = hbuf[col][kb];
            a.y  = hbuf[col][kb + 1];
            bf.x = w2[kb * WCOLS + n0 + col];
            bf.y = w2[(kb + 1) * WCOLS + n0 + col];
            c = __builtin_amdgcn_wmma_f32_16x16x4_f32(false, a, false, bf,
                                                      (short)0, c, false, false);
        }
#pragma unroll
        for (int r = 0; r < 8; ++r) {
            const int m = r + 8 * hi;
            wbuf[(e0 + m) * WCOLS + n0 + col] = c[r] * 0.125f;   // * 1/sqrt(NB)
        }
    }
}

// ---------------------------------------------------------------------------
// Kernel D: per-edge Gaunt message + scatter-add.
// Block = 256 threads = 8 wave32; ONE WAVE PER EDGE, lane = multiplicity m.
//   y[co,cx]   = sum_cf G[co,cx,cf] * attr[e,cf]        (81 vals, LDS, coop per wave)
//   msg[m,co]  = ( sum_cx y[co,cx] * x[src,m,cx] ) * ( sum_l w[e,m,l]*a_w[l,co] )
//   out[dst,m,co] += msg / denominator                   (global_atomic_add_f32)
// ---------------------------------------------------------------------------
__global__ void __launch_bounds__(256) k_message(const float* __restrict__ x,
                                                 const float* __restrict__ attr,
                                                 const int*   __restrict__ eidx,
                                                 const float* __restrict__ gbuf,
                                                 const float* __restrict__ a_w,
                                                 const float* __restrict__ wbuf,
                                                 const float* __restrict__ denom,
                                                 float* __restrict__ out) {
    __shared__ float gs[729];
    __shared__ float aws[27];
    __shared__ float ys[8][81];
    const int tid = threadIdx.x;
    for (int i = tid; i < 729; i += 256) gs[i] = gbuf[i];
    if (tid < 27) aws[tid] = a_w[tid];
    __syncthreads();

    const int wave = tid >> 5;
    const int lane = tid & 31;
    const int e    = blockIdx.x * 8 + wave;
    const int dst  = eidx[e];
    const int src  = eidx[E_EDGES + e];

    // per-lane copy of this edge's 9 filter components (L2-hot)
    float av[CF9];
#pragma unroll
    for (int c = 0; c < CF9; ++c) av[c] = attr[e * CF9 + c];

    // cooperative y[co*9+cx] into LDS: lanes j, j+32, j+64
    for (int j = lane; j < 81; j += 32) {
        const float* grow = &gs[j * 9];      // G[co,cx,:] since j = co*9+cx
        float acc = 0.f;
#pragma unroll
        for (int cf = 0; cf < CF9; ++cf) acc += grow[cf] * av[cf];
        ys[wave][j] = acc;
    }
    __syncthreads();

    const int m = lane;                       // MUL == wave size == 32
    float xv[CX9];
#pragma unroll
    for (int c = 0; c < CX9; ++c) xv[c] = x[src * XROW + m * CX9 + c];
    const float wl0 = wbuf[e * WCOLS + m * 3 + 0];
    const float wl1 = wbuf[e * WCOLS + m * 3 + 1];
    const float wl2 = wbuf[e * WCOLS + m * 3 + 2];
    const float rd  = 1.0f / denom[0];

#pragma unroll
    for (int co = 0; co < COUT9; ++co) {
        float acc = 0.f;
#pragma unroll
        for (int cxi = 0; cxi < CX9; ++cxi) acc += ys[wave][co * 9 + cxi] * xv[cxi];
        const float wm = wl0 * aws[co] + wl1 * aws[9 + co] + wl2 * aws[18 + co];
        unsafeAtomicAdd(&out[dst * XROW + m * COUT9 + co], acc * wm * rd);
    }
}

// ---------------------------------------------------------------------------
extern "C" void kernel_launch(void* const* d_in, const int* in_sizes, int n_in,
                              void* d_out, int out_size, void* d_ws, size_t ws_size,
                              hipStream_t stream) {
    const float* x     = (const float*)d_in[0];
    const float* attr  = (const float*)d_in[1];
    const float* embed = (const float*)d_in[2];
    const int*   eidx  = (const int*)  d_in[3];
    const float* Mx    = (const float*)d_in[4];   // complex64 interleaved
    const float* Mf    = (const float*)d_in[5];   // complex64 interleaved
    const float* Ms    = (const float*)d_in[6];   // complex64 interleaved
    const float* a_w   = (const float*)d_in[7];
    const float* w1    = (const float*)d_in[8];
    const float* w2    = (const float*)d_in[9];
    const float* denom = (const float*)d_in[10];
    float* out = (float*)d_out;

    float* gbuf = (float*)d_ws;                         // 729 floats
    float* wbuf = (float*)((char*)d_ws + 4096);         // E*96 floats (~6.3 MB)

    k_gaunt_G<<<81, 96, 0, stream>>>(Mx, Mf, Ms, gbuf);
    k_zero<<<(out_size + 255) / 256, 256, 0, stream>>>(out, out_size);
    k_mlp<<<E_EDGES / 16, 192, 0, stream>>>(embed, w1, w2, wbuf);
    k_message<<<E_EDGES / 8, 256, 0, stream>>>(x, attr, eidx, gbuf, a_w, wbuf,
                                               denom, out);
}